// OnsetLinspaceCell_74165495267420
// MI455X (gfx1250) — compile-verified
//
#include <hip/hip_runtime.h>
#include <stdint.h>

// Problem constants (from reference): B=64, T=8192, N=16, C=2.
#define B_   64
#define T_   8192
#define ROWF 32                 // floats per frame (16 notes * 2 ch)
#define SEG  64                 // frames per wave segment
#define SEGS (T_ / SEG)         // 128 segments per row
#define TILE_T 16               // frames per WMMA tile
#define TILES  (SEG / TILE_T)   // 4 tiles per segment
#define PADROW 36               // dwords per padded LDS row (144 B: 16B-aligned, bank-conflict-free)
#define TILE_DW (TILE_T * PADROW) // 576 dwords = 2304 B per tile buffer
#define WPB 8                   // waves per block (256 threads, wave32)

typedef float v2f __attribute__((ext_vector_type(2)));
typedef float v8f __attribute__((ext_vector_type(8)));

static __device__ __forceinline__ int imax(int a, int b) { return a > b ? a : b; }

// ---------------------------------------------------------------------------
// K1: per-wave segment scan. Async-DMA tiles into LDS, WMMA rowsum -> onset
// flag, wave-local cummax over 64 frames, write per-frame local last-index
// and per-segment max.
// ---------------------------------------------------------------------------
__global__ __launch_bounds__(256) void k1_segscan(const float* __restrict__ onsets,
                                                  int* __restrict__ lastb,
                                                  int* __restrict__ segmax) {
    __shared__ __align__(16) float smem[WPB * 2 * TILE_DW];

    const int lane = threadIdx.x & 31;
    const int wave = threadIdx.x >> 5;
    const int gw   = blockIdx.x * WPB + wave;   // global wave id, 0..8191
    const int b    = gw >> 7;                   // /128
    const int seg  = gw & (SEGS - 1);
    const int t0   = seg * SEG;                 // frame index within row

    const char* gbase = (const char*)(onsets + ((size_t)b * T_ + t0) * ROWF);
    float* wbuf = smem + wave * 2 * TILE_DW;
    const uint32_t lds0 = (uint32_t)(uintptr_t)wbuf;  // low 32 bits = LDS byte offset

    // Issue one 2KB tile (16 frames x 128B) as 4 async b128 per lane, row-padded in LDS.
    auto issue_tile = [&](int tile, int buf) {
        const char* g = gbase + tile * 2048;
        const uint32_t lb = lds0 + (uint32_t)buf * (TILE_DW * 4);
#pragma unroll
        for (int c2 = 0; c2 < 4; ++c2) {
            const uint32_t f   = (uint32_t)(c2 * 512 + lane * 16); // flat byte offset in tile
            const uint32_t lof = lb + (f >> 7) * 144u + (f & 127u); // row-padded LDS dest
            const uint64_t ga  = (uint64_t)(uintptr_t)(g + f);
            asm volatile("global_load_async_to_lds_b128 %0, %1, off"
                         :: "v"(lof), "v"(ga) : "memory");
        }
    };

    const int m    = lane & 15;   // WMMA A: lanes hold M rows
    const int half = lane >> 4;   // K-half: lanes 0-15 -> K0/K1, lanes 16-31 -> K2/K3
    int carry = -1;               // running last-onset index within segment

    issue_tile(0, 0);
#pragma unroll
    for (int tile = 0; tile < TILES; ++tile) {
        if (tile < TILES - 1) {
            issue_tile(tile + 1, (tile + 1) & 1);
            asm volatile("s_wait_asynccnt 4" ::: "memory"); // oldest tile's 4 DMAs done
        } else {
            asm volatile("s_wait_asynccnt 0" ::: "memory");
        }
        const float* buf = wbuf + (tile & 1) * TILE_DW;

        // Rowsum via matrix pipe: D[m][*] = sum over 32 interleaved floats of frame m.
        // B = ones; nonneg data + channel-broadcast mask => sum>0 <=> any(ch0)>0.
        v8f acc = {0.f, 0.f, 0.f, 0.f, 0.f, 0.f, 0.f, 0.f};
        const v2f bones = {1.0f, 1.0f};
        const float* arow = buf + m * PADROW + half * 2;
#pragma unroll
        for (int s = 0; s < 8; ++s) {
            v2f a = *(const v2f*)(arow + s * 4);  // ds_load_b64, conflict-free banks
            acc = __builtin_amdgcn_wmma_f32_16x16x4_f32(
                false, a, false, bones, (short)0, acc, false, false);
        }

        // D layout: component r = row r (lanes 0-15) / row 8+r (lanes 16-31),
        // replicated across all N columns -> every lane in a half is identical.
        const int tbase = t0 + tile * TILE_T + half * 8;
        int run = -1;
        int lastv[8];
#pragma unroll
        for (int r = 0; r < 8; ++r) {
            const int idx = (acc[r] > 0.0f) ? (tbase + r) : -1;
            run = imax(run, idx);
            lastv[r] = run;
        }
        const int lo_total = __shfl(run, 0, 32);                 // lo-half total
        const int pre = half ? imax(carry, lo_total) : carry;    // carry-in per half
#pragma unroll
        for (int r = 0; r < 8; ++r) lastv[r] = imax(lastv[r], pre);
        carry = __shfl(lastv[7], 16, 32);                        // hi-half total = new carry

        if (lane == 0 || lane == 16) {
            int* p = lastb + (size_t)b * T_ + tbase;
#pragma unroll
            for (int r = 0; r < 8; ++r) p[r] = lastv[r];
        }
    }
    if (lane == 0) segmax[gw] = carry;
}

// ---------------------------------------------------------------------------
// K2: per row, exclusive cummax over 128 segment maxima (one wave per row).
// ---------------------------------------------------------------------------
__global__ __launch_bounds__(32) void k2_carry(const int* __restrict__ segmax,
                                               int* __restrict__ segcarry) {
    const int b = blockIdx.x;
    const int lane = threadIdx.x;
    const int* src = segmax + (b << 7);
    int* dst = segcarry + (b << 7);

    int ex = -1;
    int exl[4];
#pragma unroll
    for (int j = 0; j < 4; ++j) {
        const int v = src[lane * 4 + j];
        exl[j] = ex;
        ex = imax(ex, v);
    }
    int incl = ex;
#pragma unroll
    for (int off = 1; off < 32; off <<= 1) {
        const int o = __shfl_up(incl, off, 32);
        if (lane >= off) incl = imax(incl, o);
    }
    int lex = __shfl_up(incl, 1, 32);
    if (lane == 0) lex = -1;
#pragma unroll
    for (int j = 0; j < 4; ++j) dst[lane * 4 + j] = imax(lex, exl[j]);
}

// ---------------------------------------------------------------------------
// K3: out[b,t] = t - max(local_last[b,t], segment_carry[b, t/64])
// ---------------------------------------------------------------------------
__global__ __launch_bounds__(256) void k3_finalize(const int* __restrict__ lastb,
                                                   const int* __restrict__ segcarry,
                                                   float* __restrict__ out) {
    const int i = blockIdx.x * 256 + threadIdx.x;
    const int t = i & (T_ - 1);
    const int b = i >> 13;
    const int c = segcarry[(b << 7) + (t >> 6)];
    const int l = imax(lastb[i], c);
    out[i] = (float)(t - l);
}

extern "C" void kernel_launch(void* const* d_in, const int* in_sizes, int n_in,
                              void* d_out, int out_size, void* d_ws, size_t ws_size,
                              hipStream_t stream) {
    (void)in_sizes; (void)n_in; (void)out_size; (void)ws_size;
    const float* onsets = (const float*)d_in[0];
    float* out = (float*)d_out;

    int* lastb    = (int*)d_ws;               // B*T ints (2 MB)
    int* segmax   = lastb + (size_t)B_ * T_;  // B*128 ints
    int* segcarry = segmax + B_ * SEGS;       // B*128 ints

    k1_segscan<<<(B_ * SEGS) / WPB, 256, 0, stream>>>(onsets, lastb, segmax);
    k2_carry<<<B_, 32, 0, stream>>>(segmax, segcarry);
    k3_finalize<<<(B_ * T_) / 256, 256, 0, stream>>>(lastb, segcarry, out);
}